// FeatureTokenizer_69664369541889
// MI455X (gfx1250) — compile-verified
//
#include <hip/hip_runtime.h>
#include <hip/hip_bf16.h>
#include <stdint.h>

// FeatureTokenizer: out[b,c,h] = (x[b,c]*W[c,h] + b[c,h])*(1-m[b,c]) + me[h]*m[b,c]
// B=16384, C=128, H=128. Pure bandwidth-bound broadcast op (~1.07 GB stores).
// CDNA5 path: TDM (tensor_load_to_lds) stages W/b into LDS, NT b128 stores.

#define NB_B 16384
#define NB_C 128
#define NB_H 128
#define MAT_ELEMS (NB_C * NB_H)   // 16384 floats = 64 KB per matrix

typedef float v4f __attribute__((ext_vector_type(4)));
typedef unsigned int v4u __attribute__((ext_vector_type(4)));
typedef int v4i __attribute__((ext_vector_type(4)));
typedef int v8i __attribute__((ext_vector_type(8)));

#ifndef __has_builtin
#define __has_builtin(x) 0
#endif

#if __has_builtin(__builtin_amdgcn_tensor_load_to_lds) && \
    __has_builtin(__builtin_amdgcn_s_wait_tensorcnt)
#define USE_TDM 1
#else
#define USE_TDM 0
#endif

#if USE_TDM
// Issue one TDM load of a contiguous 1-D run of f32 elements into LDS.
// D# layout per CDNA5 ISA ch.8 (group0: count/lds_addr/global_addr/type,
// group1: data_size, tensor_dim0, tile_dim0, stride). Groups 2/3 zero (<=2D).
// 6-arg builtin (clang-23): (v4u g0, v8i g1, v4i g2, v4i g3, v8i g4, i32 cpol)
__device__ __forceinline__ void tdm_load_f32_1d(unsigned lds_byte_addr,
                                                const void* gptr,
                                                unsigned nelem) {
    uint64_t ga = (uint64_t)(uintptr_t)gptr;
    v4u g0;
    g0.x = 1u;                                   // count=1, user mode, no gather
    g0.y = lds_byte_addr;                        // lds_addr [63:32]
    g0.z = (unsigned)(ga & 0xFFFFFFFFu);         // global_addr[31:0]
    g0.w = (unsigned)((ga >> 32) & 0x01FFFFFFu)  // global_addr[56:32]
         | (2u << 30);                           // type=2 ("image")
    v8i g1;
    g1[0] = (int)(2u << 16);                     // data_size=2 -> 4 bytes; mask=0
    g1[1] = (int)((nelem & 0xFFFFu) << 16);      // tensor_dim0[15:0]
    g1[2] = (int)((nelem >> 16) & 0xFFFFu)       // tensor_dim0[31:16]
          | (1 << 16);                           // tensor_dim1 = 1
    g1[3] = (int)((nelem & 0xFFFFu) << 16);      // tile_dim0 (<=65535)
    g1[4] = 1;                                   // tile_dim1=1, tile_dim2=0
    g1[5] = (int)nelem;                          // tensor_dim0_stride[31:0]
    g1[6] = 0;                                   // stride hi / dim1_stride lo
    g1[7] = 0;
    v4i z4 = {0, 0, 0, 0};
    v8i z8 = {0, 0, 0, 0, 0, 0, 0, 0};
    __builtin_amdgcn_tensor_load_to_lds(g0, g1, z4, z4, z8, 0);
}
#endif

__global__ __launch_bounds__(256)
void FeatureTokenizer_69664369541889_kernel(const float* __restrict__ x,
                                            const float* __restrict__ mask,
                                            const float* __restrict__ W,
                                            const float* __restrict__ bmat,
                                            const float* __restrict__ me,
                                            float* __restrict__ out,
                                            int numRows, int numWavesTotal) {
    __shared__ __align__(16) float smem[2 * MAT_ELEMS];  // 128 KB: W then b
    float* sW = smem;
    float* sB = smem + MAT_ELEMS;

#if USE_TDM
    if (threadIdx.x == 0) {
        unsigned lds_base = (unsigned)(uintptr_t)&smem[0];  // low 32b = LDS offset
        tdm_load_f32_1d(lds_base, W, MAT_ELEMS);
        tdm_load_f32_1d(lds_base + MAT_ELEMS * sizeof(float), bmat, MAT_ELEMS);
        __builtin_amdgcn_s_wait_tensorcnt(0);
    }
    __syncthreads();
#else
    for (int i = threadIdx.x; i < MAT_ELEMS / 4; i += blockDim.x) {
        ((v4f*)sW)[i] = ((const v4f*)W)[i];
        ((v4f*)sB)[i] = ((const v4f*)bmat)[i];
    }
    __syncthreads();
#endif

    const int lane = threadIdx.x & 31;
    // mask_embedding values for this lane's 4 h-columns: live in VGPRs forever.
    const v4f me4 = ((const v4f*)me)[lane];

    const int wavesPerBlock = blockDim.x >> 5;
    const int gwave = (int)blockIdx.x * wavesPerBlock + (int)(threadIdx.x >> 5);

    // One wave per (b,c) row; lane covers h = 4*lane .. 4*lane+3 (b128 I/O).
    for (int r = gwave; r < numRows; r += numWavesTotal) {
        const float xv = x[r];      // lane-uniform scalar (broadcast)
        const float mv = mask[r];
        const int c = r & (NB_C - 1);

        const v4f w4 = ((const v4f*)sW)[c * (NB_H / 4) + lane];  // ds_load_b128
        const v4f b4 = ((const v4f*)sB)[c * (NB_H / 4) + lane];

        const float om = 1.0f - mv;
        v4f t, o;
        t.x = fmaf(xv, w4.x, b4.x);
        t.y = fmaf(xv, w4.y, b4.y);
        t.z = fmaf(xv, w4.z, b4.z);
        t.w = fmaf(xv, w4.w, b4.w);
        o.x = fmaf(me4.x, mv, t.x * om);
        o.y = fmaf(me4.y, mv, t.y * om);
        o.z = fmaf(me4.z, mv, t.z * om);
        o.w = fmaf(me4.w, mv, t.w * om);

        // 1.07 GB written once, never re-read: non-temporal b128 store.
        __builtin_nontemporal_store(o, (v4f*)(out + ((size_t)r << 7)) + lane);
    }
}

extern "C" void kernel_launch(void* const* d_in, const int* in_sizes, int n_in,
                              void* d_out, int out_size, void* d_ws, size_t ws_size,
                              hipStream_t stream) {
    (void)in_sizes; (void)n_in; (void)out_size; (void)d_ws; (void)ws_size;
    const float* x    = (const float*)d_in[0];  // (B, C)
    const float* mask = (const float*)d_in[1];  // (B, C)
    const float* W    = (const float*)d_in[2];  // (C, H)
    const float* b    = (const float*)d_in[3];  // (C, H)
    const float* me   = (const float*)d_in[4];  // (1, H)
    float* out = (float*)d_out;                 // (B, C, H)

    const int numRows = NB_B * NB_C;            // 2,097,152 rows of 128 floats
    const int grid = 1024;                      // 128 KB LDS/block -> 2 blocks/WGP
    const int block = 256;                      // 8 waves of 32
    const int numWaves = grid * (block / 32);

    FeatureTokenizer_69664369541889_kernel<<<grid, block, 0, stream>>>(
        x, mask, W, b, me, out, numRows, numWaves);
}